// GCN_19688130085786
// MI455X (gfx1250) — compile-verified
//
#include <hip/hip_runtime.h>

#define N_NODES 100000
#define N_EDGES 1600000
#define NTILES  (N_NODES / 16)   // 6250 exact

typedef float v2f __attribute__((ext_vector_type(2)));
typedef float v8f __attribute__((ext_vector_type(8)));

// ---------------------------------------------------------------- utilities
__global__ void k_zero(float* __restrict__ p, int n) {
  int i = blockIdx.x * blockDim.x + threadIdx.x;
  if (i < n) p[i] = 0.0f;
}

__global__ void k_degree(const int* __restrict__ dst, float* __restrict__ deg) {
  int e = blockIdx.x * blockDim.x + threadIdx.x;
  if (e < N_EDGES) {
    __hip_atomic_fetch_add(&deg[dst[e]], 1.0f,
                           __ATOMIC_RELAXED, __HIP_MEMORY_SCOPE_AGENT);
  }
}

__global__ void k_deginv(float* __restrict__ deg) {
  int i = blockIdx.x * blockDim.x + threadIdx.x;
  if (i < N_NODES) {
    float d = deg[i];
    deg[i] = (d > 0.0f) ? (1.0f / d) : 0.0f;
  }
}

// Edge scatter: 8 threads per edge, each moves a float4 of the 32-dim feature
// and accumulates with hardware f32 atomics (L2-resident destination).
__global__ void k_scatter(const float* __restrict__ h,
                          const int* __restrict__ src,
                          const int* __restrict__ dst,
                          float* __restrict__ msg) {
  int t = blockIdx.x * blockDim.x + threadIdx.x;   // N_EDGES*8 = 12.8M < 2^31
  if (t >= N_EDGES * 8) return;
  int e = t >> 3;
  int c = (t & 7) * 4;
  int s = src[e];
  int d = dst[e];
  const float4 v = *(const float4*)(h + s * 32 + c);
  float* o = msg + d * 32 + c;
  __hip_atomic_fetch_add(o + 0, v.x, __ATOMIC_RELAXED, __HIP_MEMORY_SCOPE_AGENT);
  __hip_atomic_fetch_add(o + 1, v.y, __ATOMIC_RELAXED, __HIP_MEMORY_SCOPE_AGENT);
  __hip_atomic_fetch_add(o + 2, v.z, __ATOMIC_RELAXED, __HIP_MEMORY_SCOPE_AGENT);
  __hip_atomic_fetch_add(o + 3, v.w, __ATOMIC_RELAXED, __HIP_MEMORY_SCOPE_AGENT);
}

// ------------------------------------------------- SAGE layer, WMMA f32 path
// One wave computes one 16-row tile of:
//   out = sigmoid(h @ Wself + (msg * deg_inv) @ Wneigh + bias)
// using V_WMMA_F32_16X16X4_F32 with K=32 (8 steps) and OUTD/16 column tiles.
//
// A-frag (16x4 f32): lanes 0-15 -> rows M=lane, K = 4k+{0,1};
//                    lanes 16-31 -> rows M=lane-16, K = 4k+{2,3}.
// B-frag (4x16 f32): lanes 0-15 -> cols N=lane, rows K = 4k+{0,1};
//                    lanes 16-31 -> cols N=lane-16, rows K = 4k+{2,3}.
// C/D (16x16 f32):   VGPR r holds row r (lanes 0-15) / row r+8 (lanes 16-31).
template <int OUTD>
__global__ __launch_bounds__(256) void k_sage_layer(
    const float* __restrict__ h, const float* __restrict__ msg,
    const float* __restrict__ deg_inv,
    const float* __restrict__ Wself, const float* __restrict__ Wneigh,
    const float* __restrict__ bias, float* __restrict__ out) {
  constexpr int NT = OUTD / 16;
  const int lane = threadIdx.x & 31;
  const int tile = blockIdx.x * (blockDim.x >> 5) + (threadIdx.x >> 5);
  if (tile >= NTILES) return;                 // wave-uniform: EXEC stays all-1s

  const int m  = lane & 15;                    // row-within-tile / col-within-tile
  const int kh = (lane >> 4) * 2;              // K sub-offset: 0 or 2
  const int row = tile * 16 + m;
  const float di = deg_inv[row];

  v8f acc[NT];
#pragma unroll
  for (int nt = 0; nt < NT; ++nt) acc[nt] = (v8f)0.0f;

#pragma unroll
  for (int k0 = 0; k0 < 8; ++k0) {
    const int kb = k0 * 4 + kh;

    v2f aS, aN;
    const float* hp = h + row * 32 + kb;
    aS.x = hp[0];
    aS.y = hp[1];
    const float* mp = msg + row * 32 + kb;
    aN.x = mp[0] * di;
    aN.y = mp[1] * di;

    v2f bS[NT], bN[NT];
#pragma unroll
    for (int nt = 0; nt < NT; ++nt) {
      const int col = nt * 16 + m;
      bS[nt].x = Wself[(kb + 0) * OUTD + col];
      bS[nt].y = Wself[(kb + 1) * OUTD + col];
      bN[nt].x = Wneigh[(kb + 0) * OUTD + col];
      bN[nt].y = Wneigh[(kb + 1) * OUTD + col];
    }

#pragma unroll
    for (int nt = 0; nt < NT; ++nt) {
      acc[nt] = __builtin_amdgcn_wmma_f32_16x16x4_f32(
          false, aS, false, bS[nt], (short)0, acc[nt], false, false);
      acc[nt] = __builtin_amdgcn_wmma_f32_16x16x4_f32(
          false, aN, false, bN[nt], (short)0, acc[nt], false, false);
    }
  }

  const int rbase = (lane >> 4) * 8;
#pragma unroll
  for (int nt = 0; nt < NT; ++nt) {
    const float bv = bias[nt * 16 + m];
#pragma unroll
    for (int r = 0; r < 8; ++r) {
      float v = acc[nt][r] + bv;
      v = 1.0f / (1.0f + __expf(-v));          // sigmoid
      out[(tile * 16 + rbase + r) * OUTD + nt * 16 + m] = v;
    }
  }
}

// ---------------------------------------------------------------- launcher
extern "C" void kernel_launch(void* const* d_in, const int* in_sizes, int n_in,
                              void* d_out, int out_size, void* d_ws, size_t ws_size,
                              hipStream_t stream) {
  const float* inputs = (const float*)d_in[0];
  const int*   src    = (const int*)d_in[1];
  const int*   dst    = (const int*)d_in[2];
  const float* Ws1 = (const float*)d_in[3];
  const float* Wn1 = (const float*)d_in[4];
  const float* b1  = (const float*)d_in[5];
  const float* Ws2 = (const float*)d_in[6];
  const float* Wn2 = (const float*)d_in[7];
  const float* b2  = (const float*)d_in[8];
  const float* Ws3 = (const float*)d_in[9];
  const float* Wn3 = (const float*)d_in[10];
  const float* b3  = (const float*)d_in[11];
  float* out = (float*)d_out;

  // workspace layout (floats): deg_inv | msg | h1 | h2  (~38.8 MB total)
  float* ws  = (float*)d_ws;
  float* deg = ws;                       // N_NODES
  float* msg = ws + 102400;              // N_NODES*32, 16B-aligned offset
  float* h1  = msg + N_NODES * 32;
  float* h2  = h1 + N_NODES * 32;

  const int B = 256;

  // degree / inverse-degree (same for all layers)
  k_zero<<<(N_NODES + B - 1) / B, B, 0, stream>>>(deg, N_NODES);
  k_degree<<<(N_EDGES + B - 1) / B, B, 0, stream>>>(dst, deg);
  k_deginv<<<(N_NODES + B - 1) / B, B, 0, stream>>>(deg);

  const int scatterGrid = (N_EDGES * 8 + B - 1) / B;
  const int zeroGrid    = (N_NODES * 32 + B - 1) / B;
  const int tileGrid    = (NTILES + 7) / 8;          // 8 waves / block

  // layer 1: inputs -> h1  (32 -> 32)
  k_zero<<<zeroGrid, B, 0, stream>>>(msg, N_NODES * 32);
  k_scatter<<<scatterGrid, B, 0, stream>>>(inputs, src, dst, msg);
  k_sage_layer<32><<<tileGrid, B, 0, stream>>>(inputs, msg, deg, Ws1, Wn1, b1, h1);

  // layer 2: h1 -> h2  (32 -> 32)
  k_zero<<<zeroGrid, B, 0, stream>>>(msg, N_NODES * 32);
  k_scatter<<<scatterGrid, B, 0, stream>>>(h1, src, dst, msg);
  k_sage_layer<32><<<tileGrid, B, 0, stream>>>(h1, msg, deg, Ws2, Wn2, b2, h2);

  // layer 3: h2 -> out  (32 -> 16)
  k_zero<<<zeroGrid, B, 0, stream>>>(msg, N_NODES * 32);
  k_scatter<<<scatterGrid, B, 0, stream>>>(h2, src, dst, msg);
  k_sage_layer<16><<<tileGrid, B, 0, stream>>>(h2, msg, deg, Ws3, Wn3, b3, out);
}